// MultiGCNConv_22058952032941
// MI455X (gfx1250) — compile-verified
//
#include <hip/hip_runtime.h>
#include <math.h>

#define D 128

typedef __attribute__((ext_vector_type(2))) float v2f;
typedef __attribute__((ext_vector_type(8))) float v8f;

// ---------------------------------------------------------------- utilities
__global__ void fill_kernel(float* __restrict__ p, float v, int n) {
  int i = blockIdx.x * blockDim.x + threadIdx.x;
  int stride = gridDim.x * blockDim.x;
  for (; i < n; i += stride) p[i] = v;
}

__global__ void deg_kernel(const int* __restrict__ row, float* __restrict__ deg, int E) {
  int i = blockIdx.x * blockDim.x + threadIdx.x;
  if (i < E) {
    (void)__hip_atomic_fetch_add(&deg[row[i]], 1.0f, __ATOMIC_RELAXED,
                                 __HIP_MEMORY_SCOPE_AGENT);
  }
}

// norm for the E real edges only (self-loop term handled by init_kernel)
__global__ void norm_kernel(const int* __restrict__ row, const int* __restrict__ col,
                            const float* __restrict__ deg, float* __restrict__ nrm,
                            int E) {
  int i = blockIdx.x * blockDim.x + threadIdx.x;
  if (i >= E) return;
  nrm[i] = rsqrtf(deg[row[i]]) * rsqrtf(deg[col[i]]);
}

// out[i,:] = (1/deg[i]) * lin[i,:]   — the self-loop contribution seeds the
// aggregation buffer, replacing a zero-fill + N atomic self-loop edges.
__global__ void init_kernel(const float* __restrict__ lin,
                            const float* __restrict__ deg,
                            float* __restrict__ out, int n4) {
  int i = blockIdx.x * blockDim.x + threadIdx.x;
  int stride = gridDim.x * blockDim.x;
  const float4* l4 = (const float4*)lin;
  float4* o4 = (float4*)out;
  for (; i < n4; i += stride) {
    const int node = i >> 5;                 // D/4 = 32 float4 per row
    const float s = rsqrtf(deg[node]);
    const float sn = s * s;                  // dis[i]^2 == 1/deg[i]
    float4 v = l4[i];
    v.x *= sn; v.y *= sn; v.z *= sn; v.w *= sn;
    o4[i] = v;
  }
}

// ---------------------------------------------------------------- WMMA GEMM
// out[i, j] = sum_k act(h[i,k]) * W[j,k] + bias[j]   (h: NxD, W: DxD row-major)
// act = sigmoid when fused (input h is the pre-activation aggregate of the
// previous layer); each h element is read exactly once, so fusion is exact.
// 128 rows / block, 8 waves, wave w owns rows [rowBase, rowBase+16).
// Per 16x16 output tile: 32 chained v_wmma_f32_16x16x4_f32, C seeded with bias.
__global__ __launch_bounds__(256) void lin_wmma_kernel(
    const float* __restrict__ h, const float* __restrict__ W,
    const float* __restrict__ bias, float* __restrict__ out, int N,
    int apply_sigmoid) {
  __shared__ float Wl[D * D];  // 64 KB of the 320 KB/WGP LDS

  const int t = threadIdx.x;
  {  // cooperative W -> LDS stage (coalesced float4)
    const float4* Wg4 = (const float4*)W;
    float4* Wl4 = (float4*)Wl;
#pragma unroll
    for (int i = 0; i < (D * D / 4) / 256; ++i) Wl4[t + i * 256] = Wg4[t + i * 256];
  }

  const int wave  = t >> 5;
  const int lane  = t & 31;
  const int n16   = lane & 15;   // A: row-in-tile (M);  B/C: col-in-tile (N)
  const int khalf = lane >> 4;   // selects K pair {0,1} vs {2,3} (ISA A/B layout)
  const int rowBase = blockIdx.x * 128 + wave * 16;

  // Preload all 8 bias values up front (avoids a load-wait inside the tile loop)
  float bvals[8];
#pragma unroll
  for (int j = 0; j < 8; ++j) bvals[j] = bias[j * 16 + n16];

  // Preload all A fragments for this 16-row tile (keeps EXEC full: clamp rows)
  int arow = rowBase + n16;
  if (arow > N - 1) arow = N - 1;
  const float* hrow = h + (size_t)arow * D + 2 * khalf;
  v2f afrag[32];
#pragma unroll
  for (int kk = 0; kk < 32; ++kk) afrag[kk] = *(const v2f*)(hrow + 4 * kk);
  if (apply_sigmoid) {
#pragma unroll
    for (int kk = 0; kk < 32; ++kk) {
      afrag[kk].x = 1.0f / (1.0f + __expf(-afrag[kk].x));
      afrag[kk].y = 1.0f / (1.0f + __expf(-afrag[kk].y));
    }
  }

  __syncthreads();

  for (int j = 0; j < 8; ++j) {  // 8 column tiles of 16
    const int colBase = j * 16;
    v8f acc;
#pragma unroll
    for (int v = 0; v < 8; ++v) acc[v] = bvals[j];  // bias pre-seeded into C

    const float* wl = Wl + (colBase + n16) * D + 2 * khalf;
#pragma unroll
    for (int kk = 0; kk < 32; ++kk) {
      v2f bfrag = *(const v2f*)(wl + 4 * kk);   // ds_load_b64 (2addr-paired)
      acc = __builtin_amdgcn_wmma_f32_16x16x4_f32(
          /*neg_a=*/false, afrag[kk], /*neg_b=*/false, bfrag,
          /*c_mod=*/(short)0, acc, /*reuse_a=*/false, /*reuse_b=*/false);
    }

    // C/D layout: VGPR v -> M = v + 8*khalf, lane%16 -> N
#pragma unroll
    for (int v = 0; v < 8; ++v) {
      const int r = rowBase + v + 8 * khalf;
      if (r < N) out[(size_t)r * D + colBase + n16] = acc[v];
    }
  }
}

// ---------------------------------------------------------------- scatter-add
// One wave32 per real edge: each lane handles 4 floats of the 128-wide row.
__global__ __launch_bounds__(256) void scatter_kernel(
    const float* __restrict__ hin, const float* __restrict__ nrm,
    const int* __restrict__ row, const int* __restrict__ col,
    float* __restrict__ out, int E) {
  const int lane = threadIdx.x & 31;
  const int wave = threadIdx.x >> 5;
  const int e = blockIdx.x * 8 + wave;
  if (e >= E) return;

  // prefetch the gather source of the edge this wave handles 8 blocks ahead
  const int e2 = e + 64;
  if (e2 < E) {
    __builtin_prefetch(hin + (size_t)row[e2] * D + lane * 4, 0, 0);
  }

  const int r = row[e];
  const int c = col[e];
  const float nv = nrm[e];
  const float4 v = *(const float4*)(hin + (size_t)r * D + lane * 4);
  float* o = out + (size_t)c * D + lane * 4;
  (void)__hip_atomic_fetch_add(&o[0], nv * v.x, __ATOMIC_RELAXED, __HIP_MEMORY_SCOPE_AGENT);
  (void)__hip_atomic_fetch_add(&o[1], nv * v.y, __ATOMIC_RELAXED, __HIP_MEMORY_SCOPE_AGENT);
  (void)__hip_atomic_fetch_add(&o[2], nv * v.z, __ATOMIC_RELAXED, __HIP_MEMORY_SCOPE_AGENT);
  (void)__hip_atomic_fetch_add(&o[3], nv * v.w, __ATOMIC_RELAXED, __HIP_MEMORY_SCOPE_AGENT);
}

// ---------------------------------------------------------------- launcher
extern "C" void kernel_launch(void* const* d_in, const int* in_sizes, int n_in,
                              void* d_out, int out_size, void* d_ws, size_t ws_size,
                              hipStream_t stream) {
  const float* x  = (const float*)d_in[0];
  const int*   ei = (const int*)d_in[1];
  const float* Ws = (const float*)d_in[2];
  const float* bs = (const float*)d_in[3];
  float* out = (float*)d_out;

  const int N = in_sizes[0] / D;
  const int E = in_sizes[1] / 2;
  const int L = in_sizes[3] / D;
  const int* rowA = ei;       // edge_index[0] = source
  const int* colA = ei + E;   // edge_index[1] = target

  // workspace layout: [ lin output NxD | norm (E) | deg (N) ]
  float* A   = (float*)d_ws;
  float* nrm = A + (size_t)N * D;
  float* deg = nrm + (size_t)E;

  // degrees (init 1.0 = self loop contribution) and symmetric edge norm
  fill_kernel<<<256, 256, 0, stream>>>(deg, 1.0f, N);
  deg_kernel<<<(E + 255) / 256, 256, 0, stream>>>(rowA, deg, E);
  norm_kernel<<<(E + 255) / 256, 256, 0, stream>>>(rowA, colA, deg, nrm, E);

  const int linBlocks = (N + 127) / 128;
  const int scBlocks  = (E + 7) / 8;
  const int n4        = (N * D) / 4;

  const float* hin = x;
  for (int l = 0; l < L; ++l) {
    // linear (+fused sigmoid of previous layer's aggregate for l > 0)
    lin_wmma_kernel<<<linBlocks, 256, 0, stream>>>(
        hin, Ws + (size_t)l * D * D, bs + (size_t)l * D, A, N, l > 0 ? 1 : 0);
    // seed aggregate with the self-loop term: out = (1/deg) * lin
    init_kernel<<<512, 256, 0, stream>>>(A, deg, out, n4);
    // scatter-add the E real edges
    scatter_kernel<<<scBlocks, 256, 0, stream>>>(A, nrm, rowA, colA, out, E);
    hin = out;  // pre-activation aggregate feeds next layer
  }
}